// change_to_bi_65068754534652
// MI455X (gfx1250) — compile-verified
//
#include <hip/hip_runtime.h>
#include <stdint.h>

// Bit-plane extraction: out[p*8 + k] = (bit_{7-k} of floor(x[p]*255)) ? +1.0f : -1.0f
// Memory-bound streaming op: 4B in + 32B out per pixel (~347MB total @ 23.3TB/s -> ~15us).
// Strategy: B128 everywhere, CDNA5 async global->LDS double-buffered input staging
// (per-lane slots, no barriers, only s_wait_asynccnt/s_wait_dscnt), NT B128 stores
// since the 308MB output exceeds the 192MB L2 and is never re-read.

typedef uint32_t u32x4 __attribute__((ext_vector_type(4)));
typedef float    f32x4 __attribute__((ext_vector_type(4)));
// Matches the builtin's parameter type per hipcc diagnostic:
// 'int __attribute__((__vector_size__(4 * sizeof(int)))) *'
typedef int      i32x4v __attribute__((vector_size(16)));

#ifndef __has_builtin
#define __has_builtin(x) 0
#endif

#if __has_builtin(__builtin_amdgcn_global_load_async_to_lds_b128)
#define BITPLANE_ASYNC 1
#else
#define BITPLANE_ASYNC 0
#endif

#define TPB 256

// One pixel -> 8 output floats (two NT b128 stores).
// word_k = 0x3F800000 | ((~I << (24+k)) & 0x80000000), k=0 is MSB (bit 7).
__device__ __forceinline__ void emit_pixel(float f, u32x4* __restrict__ o) {
  uint32_t t = ~(uint32_t)(int)(f * 255.0f);   // trunc == floor for f >= 0
  u32x4 lo, hi;
  lo.x = 0x3F800000u | ((t << 24) & 0x80000000u);
  lo.y = 0x3F800000u | ((t << 25) & 0x80000000u);
  lo.z = 0x3F800000u | ((t << 26) & 0x80000000u);
  lo.w = 0x3F800000u | ((t << 27) & 0x80000000u);
  hi.x = 0x3F800000u | ((t << 28) & 0x80000000u);
  hi.y = 0x3F800000u | ((t << 29) & 0x80000000u);
  hi.z = 0x3F800000u | ((t << 30) & 0x80000000u);
  hi.w = 0x3F800000u | ((t << 31) & 0x80000000u);
  __builtin_nontemporal_store(lo, o);
  __builtin_nontemporal_store(hi, o + 1);
}

__global__ __launch_bounds__(TPB) void bitplane_main(const float* __restrict__ x,
                                                     uint32_t* __restrict__ out,
                                                     int n4) {
  const int lane   = (int)threadIdx.x;
  const int stride = (int)gridDim.x * TPB;
  int i = (int)blockIdx.x * TPB + lane;

#if BITPLANE_ASYNC
  // Per-lane double-buffered staging: lane L owns stage[b][L] (16B). No cross-lane
  // sharing -> no workgroup barriers, only per-wave ASYNCcnt/DScnt waits.
  __shared__ f32x4 stage[2][TPB];
  int buf = 0;
  if (i < n4) {
    __builtin_amdgcn_global_load_async_to_lds_b128(
        (i32x4v*)(x + 4 * (size_t)i), (i32x4v*)&stage[0][lane], 0, 0);
  }
  while (i < n4) {
    const int inext = i + stride;
    if (inext < n4) {
      // WAR guard: prior DS reads of stage[buf^1] must retire before async write.
      asm volatile("s_wait_dscnt 0" ::: "memory");
      __builtin_amdgcn_global_load_async_to_lds_b128(
          (i32x4v*)(x + 4 * (size_t)inext), (i32x4v*)&stage[buf ^ 1][lane], 0, 0);
      // Async loads complete in order: cnt<=1 => current buffer resident.
      asm volatile("s_wait_asynccnt 1" ::: "memory");
    } else {
      asm volatile("s_wait_asynccnt 0" ::: "memory");
    }
    f32x4 v = stage[buf][lane];                    // ds_load_b128
    u32x4* o = (u32x4*)out + (size_t)i * 8;
    emit_pixel(v.x, o);
    emit_pixel(v.y, o + 2);
    emit_pixel(v.z, o + 4);
    emit_pixel(v.w, o + 6);
    buf ^= 1;
    i = inext;
  }
#else
  for (; i < n4; i += stride) {
    f32x4 v = __builtin_nontemporal_load((const f32x4*)x + i);  // read-once input
    u32x4* o = (u32x4*)out + (size_t)i * 8;
    emit_pixel(v.x, o);
    emit_pixel(v.y, o + 2);
    emit_pixel(v.z, o + 4);
    emit_pixel(v.w, o + 6);
  }
#endif
}

// Handles n % 4 leftover pixels (not needed for 64*3*224*224, but keep it general).
__global__ void bitplane_tail(const float* __restrict__ x, uint32_t* __restrict__ out,
                              int start, int n) {
  int p = start + (int)blockIdx.x * (int)blockDim.x + (int)threadIdx.x;
  if (p < n) {
    emit_pixel(x[p], (u32x4*)out + (size_t)p * 2);
  }
}

extern "C" void kernel_launch(void* const* d_in, const int* in_sizes, int n_in,
                              void* d_out, int out_size, void* d_ws, size_t ws_size,
                              hipStream_t stream) {
  (void)n_in; (void)out_size; (void)d_ws; (void)ws_size;
  const float* x = (const float*)d_in[0];
  uint32_t* out = (uint32_t*)d_out;
  const int n  = in_sizes[0];
  const int n4 = n >> 2;

  if (n4 > 0) {
    int blocks = (n4 + TPB - 1) / TPB;
    if (blocks > 2048) blocks = 2048;   // grid-stride: >1 iter/thread so the
                                        // async double-buffer actually pipelines
    bitplane_main<<<blocks, TPB, 0, stream>>>(x, out, n4);
  }
  const int rem = n & 3;
  if (rem) {
    bitplane_tail<<<1, 64, 0, stream>>>(x, out, n4 << 2, n);
  }
}